// ImageToGraphTransformer2_60601988546909
// MI455X (gfx1250) — compile-verified
//
#include <hip/hip_runtime.h>
#include <hip/hip_bf16.h>
#include <math.h>

typedef __attribute__((ext_vector_type(2))) float v2f;
typedef __attribute__((ext_vector_type(8))) float v8f;

#define B_  32
#define C_  512
#define N_  4096   // H*W = 64*64
#define E_  1024
#define NH_ 16

// ---------------------------------------------------------------------------
// 1) x_mean[b,c] = mean over N of x[b,c,:]   (x layout [B,C,N], rows contiguous)
// ---------------------------------------------------------------------------
__global__ void mean_kernel(const float* __restrict__ x, float* __restrict__ xmean) {
    __shared__ float red[256];
    const int row = blockIdx.x;                    // b*C + c
    const float* p = x + (size_t)row * N_;
    float s = 0.f;
    for (int k = threadIdx.x; k < N_; k += 256) s += p[k];
    red[threadIdx.x] = s;
    __syncthreads();
    for (int off = 128; off > 0; off >>= 1) {
        if (threadIdx.x < off) red[threadIdx.x] += red[threadIdx.x + off];
        __syncthreads();
    }
    if (threadIdx.x == 0) xmean[row] = red[0] * (1.0f / N_);
}

// ---------------------------------------------------------------------------
// 2) pos MLP: hid = silu(xmean @ w1^T + b1); pos = 31*sigmoid(hid @ w2^T + b2)
//    one block per batch (1024 threads = 32 waves). Also writes pos_out, batch.
// ---------------------------------------------------------------------------
__global__ void mlp_kernel(const float* __restrict__ xmean,
                           const float* __restrict__ w1, const float* __restrict__ b1,
                           const float* __restrict__ w2, const float* __restrict__ b2,
                           float* __restrict__ pos_ws, float* __restrict__ pos_out,
                           float* __restrict__ batch_out) {
    __shared__ float xm[C_];
    __shared__ float hid[E_];
    const int b = blockIdx.x, t = threadIdx.x;
    if (t < C_) xm[t] = xmean[b * C_ + t];
    __syncthreads();

    float acc = b1[t];
    const float* wr = w1 + (size_t)t * C_;
    #pragma unroll 4
    for (int c = 0; c < C_; ++c) acc = fmaf(xm[c], wr[c], acc);
    hid[t] = acc / (1.0f + expf(-acc));            // SiLU
    __syncthreads();

    if (t < 2 * NH_) {                             // 32 logits
        float o = b2[t];
        const float* w = w2 + (size_t)t * E_;
        #pragma unroll 4
        for (int e = 0; e < E_; ++e) o = fmaf(hid[e], w[e], o);
        const float p = 31.0f / (1.0f + expf(-o)); // 31*sigmoid
        pos_ws[b * 32 + t]  = p;
        pos_out[b * 32 + t] = p;
    }
    if (t < NH_) batch_out[b * NH_ + t] = (float)b;
}

// ---------------------------------------------------------------------------
// 3) attention[b,n,h] = norm*exp(-0.5*((i-px)^2+(j-py)^2)/sigma^2); pen[b,h]=sum_n
//    grid = (N/256, B), block = 256 threads; one (b,n) per thread, all 16 heads.
// ---------------------------------------------------------------------------
__global__ void att_kernel(const float* __restrict__ pos,
                           float* __restrict__ att_out, float* __restrict__ pen) {
    __shared__ float ps[2 * NH_];
    __shared__ float red[NH_ * 256];
    const int b = blockIdx.y;
    const int n = blockIdx.x * 256 + threadIdx.x;
    if (threadIdx.x < 2 * NH_) ps[threadIdx.x] = pos[b * 32 + threadIdx.x];
    __syncthreads();

    const float fi = (float)(n >> 6);              // row index  (grid comp 0)
    const float fj = (float)(n & 63);              // col index  (grid comp 1)
    const float norm = 0.01768387f;                // 1/(2*pi*9)
    float a[NH_];
    #pragma unroll
    for (int h = 0; h < NH_; ++h) {
        const float dx = fi - ps[2 * h];
        const float dy = fj - ps[2 * h + 1];
        const float v = norm * expf(-(dx * dx + dy * dy) * (0.5f / 9.0f));
        a[h] = v;
        red[h * 256 + threadIdx.x] = v;
    }
    float4* outp = (float4*)(att_out + ((size_t)b * N_ + n) * NH_);
    outp[0] = make_float4(a[0],  a[1],  a[2],  a[3]);
    outp[1] = make_float4(a[4],  a[5],  a[6],  a[7]);
    outp[2] = make_float4(a[8],  a[9],  a[10], a[11]);
    outp[3] = make_float4(a[12], a[13], a[14], a[15]);
    __syncthreads();
    if (threadIdx.x < NH_) {
        float s = 0.f;
        for (int k = 0; k < 256; ++k) s += red[threadIdx.x * 256 + k];
        atomicAdd(&pen[b * NH_ + threadIdx.x], s);
    }
}

// ---------------------------------------------------------------------------
// 4) y[b,h,c] = sum_n att[b,n,h] * x[b,c,n]  via V_WMMA_F32_16X16X4_F32.
//    M=heads(16), N=c-tile(16), K=n. 4 waves/block, one c-tile per wave.
//    A 16x4 fp32 layout (ISA 7.12.2): lanes 0-15 -> K=k+{0,1}, lanes 16-31 -> K=k+{2,3}.
// ---------------------------------------------------------------------------
__global__ void ygemm_kernel(const float* __restrict__ att, const float* __restrict__ x,
                             float* __restrict__ y) {
    const int b    = blockIdx.y;
    const int wave = threadIdx.x >> 5;
    const int lane = threadIdx.x & 31;
    const int ct   = blockIdx.x * 4 + wave;        // channel tile 0..31
    const int m    = lane & 15;
    const int half = lane >> 4;
    const int kb   = 2 * half;

    const float* A  = att + (size_t)b * N_ * NH_;                      // A[k][m] = att[b,k,m]
    const float* Xb = x + (size_t)b * C_ * N_ + (size_t)(ct * 16 + m) * N_;

    v8f acc = {};
    #pragma unroll 4
    for (int k = 0; k < N_; k += 4) {
        v2f a, bb;
        a.x = A[(k + kb)     * NH_ + m];
        a.y = A[(k + kb + 1) * NH_ + m];
        bb  = *(const v2f*)(Xb + k + kb);
        acc = __builtin_amdgcn_wmma_f32_16x16x4_f32(false, a, false, bb,
                                                    (short)0, acc, false, false);
    }
    float* Yb = y + (size_t)b * NH_ * C_ + ct * 16 + m;
    #pragma unroll
    for (int r = 0; r < 8; ++r) {
        const int row = r + 8 * half;              // head index
        Yb[row * C_] = acc[r];
    }
}

// ---------------------------------------------------------------------------
// 5) values[row,e] = sum_c y[row,c]*wv[e,c] + pen[row]*bv[e];   row = b*NH+h.
//    M=512 (row tiles), N=1024 (e tiles), K=512. 4 waves/block over e-tiles.
// ---------------------------------------------------------------------------
__global__ void vgemm_kernel(const float* __restrict__ y, const float* __restrict__ wv,
                             const float* __restrict__ bv, const float* __restrict__ pen,
                             float* __restrict__ values) {
    const int wave = threadIdx.x >> 5;
    const int lane = threadIdx.x & 31;
    const int et   = blockIdx.x * 4 + wave;        // e tile 0..63
    const int mt   = blockIdx.y;                   // row tile 0..31
    const int m    = lane & 15;
    const int half = lane >> 4;
    const int kb   = 2 * half;

    const float* Arow = y  + (size_t)(mt * 16 + m) * C_;   // A[mrow][k]
    const float* Brow = wv + (size_t)(et * 16 + m) * C_;   // B[k][e=m] = wv[e][k]

    v8f acc = {};
    #pragma unroll 4
    for (int k = 0; k < C_; k += 4) {
        v2f a  = *(const v2f*)(Arow + k + kb);
        v2f bb = *(const v2f*)(Brow + k + kb);
        acc = __builtin_amdgcn_wmma_f32_16x16x4_f32(false, a, false, bb,
                                                    (short)0, acc, false, false);
    }
    const int col = et * 16 + m;
    const float bvc = bv[col];
    #pragma unroll
    for (int r = 0; r < 8; ++r) {
        const int row = mt * 16 + r + 8 * half;
        values[(size_t)row * E_ + col] = acc[r] + pen[row] * bvc;
    }
}

// ---------------------------------------------------------------------------
// 6) penalty = mean((pen - 1)^2) over 512 values
// ---------------------------------------------------------------------------
__global__ void penalty_kernel(const float* __restrict__ pen, float* __restrict__ out) {
    __shared__ float red[512];
    const float d = pen[threadIdx.x] - 1.0f;
    red[threadIdx.x] = d * d;
    __syncthreads();
    for (int off = 256; off > 0; off >>= 1) {
        if (threadIdx.x < off) red[threadIdx.x] += red[threadIdx.x + off];
        __syncthreads();
    }
    if (threadIdx.x == 0) out[0] = red[0] * (1.0f / 512.0f);
}

// ---------------------------------------------------------------------------
extern "C" void kernel_launch(void* const* d_in, const int* in_sizes, int n_in,
                              void* d_out, int out_size, void* d_ws, size_t ws_size,
                              hipStream_t stream) {
    const float* x  = (const float*)d_in[0];
    const float* w1 = (const float*)d_in[1];
    const float* b1 = (const float*)d_in[2];
    const float* w2 = (const float*)d_in[3];
    const float* b2 = (const float*)d_in[4];
    const float* wv = (const float*)d_in[5];
    const float* bv = (const float*)d_in[6];

    float* out       = (float*)d_out;
    float* values    = out;                        // [512,1024] = 524288
    float* pos_out   = out + 524288;               // [512,2]    = 1024
    float* batch_out = out + 525312;               // [512]      = 512
    float* attention = out + 525824;               // [32,4096,16] = 2097152
    float* penalty   = out + 2622976;              // scalar

    float* ws    = (float*)d_ws;
    float* xmean = ws;                             // 16384 floats
    float* pos   = ws + 16384;                     // 1024
    float* pen   = ws + 17408;                     // 512
    float* y     = ws + 17920;                     // 262144 (B*NH*C)

    hipMemsetAsync(pen, 0, 512 * sizeof(float), stream);

    mean_kernel   <<<B_ * C_,            256, 0, stream>>>(x, xmean);
    mlp_kernel    <<<B_,                 1024, 0, stream>>>(xmean, w1, b1, w2, b2,
                                                            pos, pos_out, batch_out);
    att_kernel    <<<dim3(N_ / 256, B_), 256, 0, stream>>>(pos, attention, pen);
    ygemm_kernel  <<<dim3(C_ / 64, B_),  128, 0, stream>>>(attention, x, y);
    vgemm_kernel  <<<dim3(E_ / 64, 32),  128, 0, stream>>>(y, wv, bv, pen, values);
    penalty_kernel<<<1,                  512, 0, stream>>>(pen, penalty);
}